// Loss_Functions_19945828123268
// MI455X (gfx1250) — compile-verified
//
#include <hip/hip_runtime.h>

// ---------------------------------------------------------------------------
// Dice loss for pred(8,19,512,1024) f32, target(8,2,512,1024) f32.
// Bandwidth-bound: 352 MB mandatory traffic -> ~15us floor at 23.3 TB/s.
// Single pass over pred (pixel-major, all 19 channels per pixel, coalesced
// b128 loads); label-dependent terms folded in with compare+select so no
// gather pass is needed. Block-level reduction finishes with
// V_WMMA_F32_16X16X4_F32 (A = 4 partials/value per documented 16x4 f32 A
// layout, B = all-ones which is layout independent, D = totals per the
// documented C/D layout).
// ---------------------------------------------------------------------------

#define BB     8
#define CCH    19
#define HH     512
#define WW     1024
#define HW     (HH * WW)          // 524288 = 1<<19
#define NPIX   (BB * HW)          // 4194304
#define TILE   1024               // pixels per block-iteration (256 thr * 4)
#define NTILES (NPIX / TILE)      // 4096 (512 tiles per batch image)
#define TPB    256
#define NBLK   1024
#define NV     64                 // 57 live values padded to 64
#define GSTRIDE 32                // float stride per accumulator (atomic spread)

typedef __attribute__((ext_vector_type(2))) float v2f;
typedef __attribute__((ext_vector_type(8))) float v8f;

// Sum x across each group of 8 consecutive lanes (wave32) via ds_swizzle_b32,
// group-of-32 mode: and=0x1f, or=0, xor in bits [14:10].
__device__ __forceinline__ float grp8_sum(float x) {
  x += __int_as_float(__builtin_amdgcn_ds_swizzle(__float_as_int(x), 0x041F)); // xor 1
  x += __int_as_float(__builtin_amdgcn_ds_swizzle(__float_as_int(x), 0x081F)); // xor 2
  x += __int_as_float(__builtin_amdgcn_ds_swizzle(__float_as_int(x), 0x101F)); // xor 4
  return x;
}

__global__ void zero_ws_kernel(float* __restrict__ g, int n) {
  int i = blockIdx.x * blockDim.x + threadIdx.x;
  if (i < n) g[i] = 0.0f;
}

__global__ __launch_bounds__(TPB) void dice_accum_kernel(
    const float* __restrict__ pred, const float* __restrict__ target,
    float* __restrict__ gacc) {
  // 32 partial rows (8 waves x 4 lane-groups) x 64 padded values.
  __shared__ float part[32][NV + 4];

  float interA[CCH], tsumA[CCH], psumA[CCH];
#pragma unroll
  for (int c = 0; c < CCH; ++c) { interA[c] = 0.f; tsumA[c] = 0.f; psumA[c] = 0.f; }

  const int tid = threadIdx.x;

  for (int tile = blockIdx.x; tile < NTILES; tile += gridDim.x) {
    const int b = tile >> 9;                               // 512 tiles per image
    const size_t off = ((size_t)(tile & 511)) * TILE + (size_t)tid * 4;
    const size_t tbase = ((size_t)b << 20);                // b * 2 * HW

    const float4 lf = *(const float4*)(target + tbase + off);          // labels (as float)
    const float4 wf = *(const float4*)(target + tbase + (size_t)HW + off); // weights
    const int l0 = (int)lf.x, l1 = (int)lf.y, l2 = (int)lf.z, l3 = (int)lf.w;

    // gfx1250 prefetch of the next tile's labels (global_prefetch_b8).
    if (tile + (int)gridDim.x < NTILES) {
      const int nt = tile + (int)gridDim.x;
      const size_t noff = ((size_t)(nt & 511)) * TILE + (size_t)tid * 4;
      __builtin_prefetch(target + (((size_t)(nt >> 9)) << 20) + noff, 0, 0);
    }

#pragma unroll
    for (int c = 0; c < CCH; ++c) {
      const float4 p4 = *(const float4*)(pred + (((size_t)(b * CCH + c)) << 19) + off);
      const float q0 = p4.x * wf.x, q1 = p4.y * wf.y;
      const float q2 = p4.z * wf.z, q3 = p4.w * wf.w;
      psumA[c]  += (q0 + q1) + (q2 + q3);
      interA[c] += (l0 == c ? q0 : 0.f) + (l1 == c ? q1 : 0.f)
                 + (l2 == c ? q2 : 0.f) + (l3 == c ? q3 : 0.f);
      tsumA[c]  += (l0 == c ? wf.x : 0.f) + (l1 == c ? wf.y : 0.f)
                 + (l2 == c ? wf.z : 0.f) + (l3 == c ? wf.w : 0.f);
    }
  }

  // Reduce each accumulator over 8-lane groups, one row per group.
#pragma unroll
  for (int c = 0; c < CCH; ++c) {
    interA[c] = grp8_sum(interA[c]);
    tsumA[c]  = grp8_sum(tsumA[c]);
    psumA[c]  = grp8_sum(psumA[c]);
  }
  if ((tid & 7) == 0) {
    const int r = tid >> 3;  // wave*4 + group, 0..31
#pragma unroll
    for (int c = 0; c < CCH; ++c) {
      part[r][c]      = interA[c];
      part[r][19 + c] = tsumA[c];
      part[r][38 + c] = psumA[c];
    }
#pragma unroll
    for (int v = 57; v < NV; ++v) part[r][v] = 0.f;
  }
  __syncthreads();

  if (tid < 32) {           // wave 0 only, EXEC all-ones inside (uniform branch)
    const int L = tid;
    // Fold 32 partial rows -> 4 rows (K dimension for the WMMA).
#pragma unroll
    for (int v0 = 0; v0 < 2; ++v0) {
      const int v = L + 32 * v0;
      float s0 = 0.f, s1 = 0.f, s2 = 0.f, s3 = 0.f;
#pragma unroll
      for (int r = 0; r < 8; ++r) {
        s0 += part[r][v];       s1 += part[8 + r][v];
        s2 += part[16 + r][v];  s3 += part[24 + r][v];
      }
      part[0][v] = s0; part[1][v] = s1; part[2][v] = s2; part[3][v] = s3;
    }

    const int col = L & 15, hi = L >> 4;
    v2f bones; bones[0] = 1.0f; bones[1] = 1.0f;   // B all-ones: layout independent
#pragma unroll
    for (int vb = 0; vb < 4; ++vb) {
      // A 16x4 f32 layout: lanes 0-15 hold K=0,1; lanes 16-31 hold K=2,3.
      v2f a;
      a[0] = part[2 * hi][vb * 16 + col];
      a[1] = part[2 * hi + 1][vb * 16 + col];
      v8f c0 = {};
      v8f d = __builtin_amdgcn_wmma_f32_16x16x4_f32(
          false, a, false, bones, (short)0, c0, false, false);
      // D layout: VGPR j, lanes 0-15 -> M=j; lanes 16-31 -> M=j+8.
      if ((L & 15) == 0) {
        const int base = vb * 16 + hi * 8;
#pragma unroll
        for (int j = 0; j < 8; ++j) {
          const int v = base + j;
          if (v < 57) atomicAdd(gacc + (size_t)v * GSTRIDE, d[j]);
        }
      }
    }
  }
}

__global__ void finalize_kernel(const float* __restrict__ g, float* __restrict__ out) {
  if (threadIdx.x == 0 && blockIdx.x == 0) {
    float loss = 0.0f;
#pragma unroll
    for (int c = 0; c < CCH; ++c) {
      const float inter = g[(size_t)c * GSTRIDE];
      const float tsum  = g[(size_t)(19 + c) * GSTRIDE];
      const float psum  = g[(size_t)(38 + c) * GSTRIDE];
      const float dice  = (2.0f * inter + 1.0f) / (psum + tsum + 1.0f);
      loss += 1.0f - dice;
    }
    out[0] = loss / (float)CCH;
  }
}

extern "C" void kernel_launch(void* const* d_in, const int* in_sizes, int n_in,
                              void* d_out, int out_size, void* d_ws, size_t ws_size,
                              hipStream_t stream) {
  const float* pred   = (const float*)d_in[0];
  const float* target = (const float*)d_in[1];
  float* out  = (float*)d_out;
  float* gacc = (float*)d_ws;

  const int nzero = 57 * GSTRIDE;  // 1824 floats
  zero_ws_kernel<<<(nzero + 255) / 256, 256, 0, stream>>>(gacc, nzero);
  dice_accum_kernel<<<NBLK, TPB, 0, stream>>>(pred, target, gacc);
  finalize_kernel<<<1, 1, 0, stream>>>(gacc, out);
}